// TensorODEBLOCK_43714177139083
// MI455X (gfx1250) — compile-verified
//
#include <hip/hip_runtime.h>

// ---------------------------------------------------------------------------
// Batched adaptive RK45 for dY/dt = Y @ C^T, fully LDS-resident per row-tile.
//   B=512 rows, state dim DT=512, DIN=128, DOUT=256, 64 steps.
// Each block owns 16 ODE rows; the whole solve runs with zero global sync.
// GEMMs use V_WMMA_F32_16X16X4_F32 (gfx1250 f32 matrix path).
// RK stage combinations are computed ONCE per block (wave-per-row pass) into
// the Ynew buffer, which doubles as the A-staging buffer for every f-eval,
// keeping the GEMM inner loop lean: 1x ds_load_b64 + 2x global b64 + 2x WMMA.
// ---------------------------------------------------------------------------

#define ROWS    16
#define DTN     512
#define DINN    128
#define DOUTN   256
#define BATCH   512
#define STRIDE  516                      // 512 + 4 pad floats: spreads LDS banks
#define NBUF    9                        // Y, K0..K6, A-staging/Ynew
#define STATF   128
#define SMEM_FLOATS (NBUF * ROWS * STRIDE + STATF)
#define SMEM_BYTES  (SMEM_FLOATS * 4)

#define RTOL 1.0e-3f
#define ATOL 1.0e-6f
#define TINY 1.0e-30f

typedef __attribute__((ext_vector_type(2))) float v2f;
typedef __attribute__((ext_vector_type(8))) float v8f;

#define BUFO(i) ((i) * ROWS * STRIDE)
#define ABUF 8                            // staging buffer index (also Ynew)

__device__ __forceinline__ v8f wmma4(v2f a, v2f b, v8f c) {
  // 8 args: (neg_a, A, neg_b, B, c_mod, C, reuse_a, reuse_b)
  return __builtin_amdgcn_wmma_f32_16x16x4_f32(false, a, false, b, (short)0, c,
                                               false, false);
}

// Dormand-Prince stage coefficients
__constant__ float cA1[1] = {0.2f};
__constant__ float cA2[2] = {0.075f, 0.225f};
__constant__ float cA3[3] = {44.f / 45.f, -56.f / 15.f, 32.f / 9.f};
__constant__ float cA4[4] = {19372.f / 6561.f, -25360.f / 2187.f,
                             64448.f / 6561.f, -212.f / 729.f};
__constant__ float cA5[5] = {9017.f / 3168.f, -355.f / 33.f, 46732.f / 5247.f,
                             49.f / 176.f, -5103.f / 18656.f};
__constant__ float cOne[1] = {1.0f};

#define B0 (35.f / 384.f)
#define B2 (500.f / 1113.f)
#define B3 (125.f / 192.f)
#define B4 (-2187.f / 6784.f)
#define B5 (11.f / 84.f)
#define E0 (71.f / 57600.f)
#define E2 (-71.f / 16695.f)
#define E3 (71.f / 1920.f)
#define E4 (-17253.f / 339200.f)
#define E5 (22.f / 525.f)
#define E6 (-1.f / 40.f)

// A-staging: BUFO(8)[m][k] = Y[m][k] + hrow[m] * sum_j coef[j]*K_j[m][k]
// One wave per row -> combination computed exactly once per block.
template <int NC>
__device__ __forceinline__ void comb_pass(float* sm, const float* coef,
                                          const float* hrow, int lane,
                                          int wave) {
  const int m = wave;
  float cj[NC];
  const float hm = hrow[m];
#pragma unroll
  for (int j = 0; j < NC; ++j) cj[j] = hm * coef[j];
#pragma unroll
  for (int it = 0; it < 16; ++it) {
    const size_t idx = (size_t)m * STRIDE + lane + 32 * it;
    float a = sm[idx];
#pragma unroll
    for (int j = 0; j < NC; ++j) a = fmaf(cj[j], sm[BUFO(1 + j) + idx], a);
    sm[BUFO(ABUF) + idx] = a;
  }
}

// K[dst](16x512) = A[aBuf] @ C^T ; lean inner loop under the WMMA pipe.
__device__ __forceinline__ void eval_gemm(float* sm,
                                          const float* __restrict__ Cg,
                                          int dstBuf, int aBuf, int lane,
                                          int wave) {
  const int m = lane & 15;          // A row / B column within tile
  const int hi = lane >> 4;         // lane half
  const int khalf = hi << 1;        // K sub-offset 0 or 2
  const int n0 = wave * 32 + m;     // this wave owns 32 output columns
  const int n1 = n0 + 16;

  const float* A = sm + (size_t)aBuf * ROWS * STRIDE + (size_t)m * STRIDE;
  const float* Crow0 = Cg + (size_t)n0 * DTN;
  const float* Crow1 = Cg + (size_t)n1 * DTN;
  v8f acc0 = {};
  v8f acc1 = {};

#pragma unroll 8
  for (int kk = 0; kk < DTN; kk += 4) {
    const int kb = kk + khalf;
    v2f a = *(const v2f*)(A + kb);  // 16x4 f32 A layout: lane m, K pair kb,kb+1
    // B = C^T slice: B[kb_rel][n] = C[n][kb]; contiguous float2 per lane
    v2f b0 = *(const v2f*)(Crow0 + kb);
    v2f b1 = *(const v2f*)(Crow1 + kb);
    acc0 = wmma4(a, b0, acc0);
    acc1 = wmma4(a, b1, acc1);
  }

  // C/D layout: VGPR r, lane half hi -> row r + 8*hi, col = lane&15
  float* D = sm + (size_t)dstBuf * ROWS * STRIDE;
#pragma unroll
  for (int r = 0; r < 8; ++r) {
    const int mr = r + (hi << 3);
    D[(size_t)mr * STRIDE + n0] = acc0[r];
    D[(size_t)mr * STRIDE + n1] = acc1[r];
  }
}

__global__ __launch_bounds__(512) void TensorODE_rk45_kernel(
    const float* __restrict__ x, const float* __restrict__ P,
    const float* __restrict__ Cm, const float* __restrict__ F,
    float* __restrict__ out) {
  extern __shared__ float sm[];
  const int tid = threadIdx.x;
  const int lane = tid & 31;
  const int wave = tid >> 5;  // 16 waves; elementwise phases: wave == ODE row
  const int rowbase = blockIdx.x * ROWS;

  float* stat = sm + NBUF * ROWS * STRIDE;
  float* t_ = stat + 0;
  float* h_ = stat + 16;
  float* hs_ = stat + 32;
  float* rej_ = stat + 48;
  float* acc_ = stat + 64;
  float* h0_ = stat + 80;
  float* d1_ = stat + 96;

  // ---- Phase 1: Y = x @ P   (16x128 @ 128x512) ----
  {
    const int m = lane & 15;
    const int hi = lane >> 4;
    const int khalf = hi << 1;
    const int n0 = wave * 32 + m, n1 = n0 + 16;
    v8f a0 = {}, a1 = {};
#pragma unroll 4
    for (int kk = 0; kk < DINN; kk += 4) {
      const int kb = kk + khalf;
      v2f a = *(const v2f*)(x + (size_t)(rowbase + m) * DINN + kb);
      v2f b0, b1;
      b0.x = P[(size_t)kb * DTN + n0];
      b0.y = P[(size_t)(kb + 1) * DTN + n0];
      b1.x = P[(size_t)kb * DTN + n1];
      b1.y = P[(size_t)(kb + 1) * DTN + n1];
      a0 = wmma4(a, b0, a0);
      a1 = wmma4(a, b1, a1);
    }
#pragma unroll
    for (int r = 0; r < 8; ++r) {
      const int mr = r + (hi << 3);
      sm[(size_t)mr * STRIDE + n0] = a0[r];
      sm[(size_t)mr * STRIDE + n1] = a1[r];
    }
  }
  __syncthreads();

  // ---- Phase 2: K0 = f(Y) ----
  eval_gemm(sm, Cm, 1, 0, lane, wave);
  __syncthreads();

  // ---- Phase 3: d0,d1 -> h0 (one wave per row) ----
  {
    const int m = wave;
    float s0 = 0.f, s1 = 0.f;
#pragma unroll
    for (int it = 0; it < 16; ++it) {
      const int k = lane + 32 * it;
      const float y = sm[(size_t)m * STRIDE + k];
      const float f0 = sm[BUFO(1) + (size_t)m * STRIDE + k];
      const float sc = ATOL + fabsf(y) * RTOL;
      const float a = y / sc, b = f0 / sc;
      s0 += a * a;
      s1 += b * b;
    }
    for (int o = 16; o > 0; o >>= 1) {
      s0 += __shfl_xor(s0, o, 32);
      s1 += __shfl_xor(s1, o, 32);
    }
    if (lane == 0) {
      const float d0 = sqrtf(fmaxf(s0 * (1.f / DTN), TINY));
      const float d1 = sqrtf(fmaxf(s1 * (1.f / DTN), TINY));
      h0_[m] = (d0 < 1e-5f || d1 < 1e-5f) ? 1e-6f : 0.01f * d0 / d1;
      d1_[m] = d1;
    }
  }
  __syncthreads();

  // ---- Phase 4: K1 = f(Y + h0*K0) ----
  comb_pass<1>(sm, cOne, h0_, lane, wave);
  __syncthreads();
  eval_gemm(sm, Cm, 2, ABUF, lane, wave);
  __syncthreads();

  // ---- Phase 5: d2 -> initial h ----
  {
    const int m = wave;
    float s2 = 0.f;
#pragma unroll
    for (int it = 0; it < 16; ++it) {
      const int k = lane + 32 * it;
      const float y = sm[(size_t)m * STRIDE + k];
      const float f0 = sm[BUFO(1) + (size_t)m * STRIDE + k];
      const float f1 = sm[BUFO(2) + (size_t)m * STRIDE + k];
      const float sc = ATOL + fabsf(y) * RTOL;
      const float d = (f1 - f0) / sc;
      s2 += d * d;
    }
    for (int o = 16; o > 0; o >>= 1) s2 += __shfl_xor(s2, o, 32);
    if (lane == 0) {
      const float h0 = h0_[m];
      const float d2 = sqrtf(fmaxf(s2 * (1.f / DTN), TINY)) / h0;
      const float d1 = d1_[m];
      const float maxd = fmaxf(d1, d2);
      const float h1 = (maxd <= 1e-15f)
                           ? fmaxf(1e-6f, h0 * 1e-3f)
                           : powf(0.01f / fmaxf(maxd, 1e-15f), 0.2f);
      h_[m] = fminf(fminf(100.f * h0, h1), 1.0f);
      t_[m] = 0.f;
      rej_[m] = 0.f;
    }
  }
  __syncthreads();

  // ---- Phase 6: 64 adaptive steps ----
  for (int step = 0; step < 64; ++step) {
    if (tid < 16) hs_[tid] = fminf(h_[tid], 1.0f - t_[tid]);
    __syncthreads();

    comb_pass<1>(sm, cA1, hs_, lane, wave); __syncthreads();
    eval_gemm(sm, Cm, 2, ABUF, lane, wave); __syncthreads();
    comb_pass<2>(sm, cA2, hs_, lane, wave); __syncthreads();
    eval_gemm(sm, Cm, 3, ABUF, lane, wave); __syncthreads();
    comb_pass<3>(sm, cA3, hs_, lane, wave); __syncthreads();
    eval_gemm(sm, Cm, 4, ABUF, lane, wave); __syncthreads();
    comb_pass<4>(sm, cA4, hs_, lane, wave); __syncthreads();
    eval_gemm(sm, Cm, 5, ABUF, lane, wave); __syncthreads();
    comb_pass<5>(sm, cA5, hs_, lane, wave); __syncthreads();
    eval_gemm(sm, Cm, 6, ABUF, lane, wave); __syncthreads();

    // Ynew = Y + hs*(B0*K0 + B2*K2 + B3*K3 + B4*K4 + B5*K5) -> staging buffer
    {
      const int m = wave;
      const float hm = hs_[m];
#pragma unroll
      for (int it = 0; it < 16; ++it) {
        const size_t idx = (size_t)m * STRIDE + lane + 32 * it;
        const float yn =
            sm[idx] + hm * (B0 * sm[BUFO(1) + idx] + B2 * sm[BUFO(3) + idx] +
                            B3 * sm[BUFO(4) + idx] + B4 * sm[BUFO(5) + idx] +
                            B5 * sm[BUFO(6) + idx]);
        sm[BUFO(ABUF) + idx] = yn;
      }
    }
    __syncthreads();

    // K6 = f(Ynew)
    eval_gemm(sm, Cm, 7, ABUF, lane, wave);
    __syncthreads();

    // error norm + per-row step control (one wave per row)
    {
      const int m = wave;
      const float hm = hs_[m];
      float s = 0.f;
#pragma unroll
      for (int it = 0; it < 16; ++it) {
        const size_t idx = (size_t)m * STRIDE + lane + 32 * it;
        const float y = sm[idx];
        const float yn = sm[BUFO(ABUF) + idx];
        const float err =
            hm * (E0 * sm[BUFO(1) + idx] + E2 * sm[BUFO(3) + idx] +
                  E3 * sm[BUFO(4) + idx] + E4 * sm[BUFO(5) + idx] +
                  E5 * sm[BUFO(6) + idx] + E6 * sm[BUFO(7) + idx]);
        const float sc = ATOL + RTOL * fmaxf(fabsf(y), fabsf(yn));
        const float r = err / sc;
        s += r * r;
      }
      for (int o = 16; o > 0; o >>= 1) s += __shfl_xor(s, o, 32);
      if (lane == 0) {
        const float en = sqrtf(fmaxf(s * (1.f / DTN), TINY));
        const bool acc = en < 1.0f;
        const float pw = 0.9f * powf(en, -0.2f);
        float fa = fminf(10.0f, pw);
        if (rej_[m] != 0.f) fa = fminf(fa, 1.0f);
        const float fac = acc ? fa : fmaxf(0.2f, pw);
        const float hsv = hs_[m];
        if (acc) t_[m] += hsv;
        h_[m] = hsv * fac;
        rej_[m] = acc ? 0.f : 1.f;
        acc_[m] = acc ? 1.f : 0.f;
      }
    }
    __syncthreads();

    // accept: Y <- Ynew, K0 <- K6
    {
      const int m = wave;
      if (acc_[m] != 0.f) {
#pragma unroll
        for (int it = 0; it < 16; ++it) {
          const size_t idx = (size_t)m * STRIDE + lane + 32 * it;
          sm[idx] = sm[BUFO(ABUF) + idx];
          sm[BUFO(1) + idx] = sm[BUFO(7) + idx];
        }
      }
    }
    __syncthreads();
  }

  // ---- Phase 7: out = Y @ F   (16x512 @ 512x256), 1 tile per wave ----
  {
    const int m = lane & 15;
    const int hi = lane >> 4;
    const int khalf = hi << 1;
    const int n = wave * 16 + m;
    v8f acc = {};
#pragma unroll 4
    for (int kk = 0; kk < DTN; kk += 4) {
      const int kb = kk + khalf;
      v2f a = *(const v2f*)(sm + (size_t)m * STRIDE + kb);
      v2f b;
      b.x = F[(size_t)kb * DOUTN + n];
      b.y = F[(size_t)(kb + 1) * DOUTN + n];
      acc = wmma4(a, b, acc);
    }
#pragma unroll
    for (int r = 0; r < 8; ++r) {
      const int mr = r + (hi << 3);
      out[(size_t)(rowbase + mr) * DOUTN + n] = acc[r];
    }
  }
}

extern "C" void kernel_launch(void* const* d_in, const int* in_sizes, int n_in,
                              void* d_out, int out_size, void* d_ws,
                              size_t ws_size, hipStream_t stream) {
  const float* x = (const float*)d_in[0];
  const float* P = (const float*)d_in[1];
  const float* C = (const float*)d_in[2];
  const float* F = (const float*)d_in[3];
  float* out = (float*)d_out;
  (void)in_sizes; (void)n_in; (void)out_size; (void)d_ws; (void)ws_size;

  // 290.75 KB dynamic LDS per block (<= 320 KB WGP LDS); raise the cap.
  // Host-side attribute set, not a stream op -> safe under graph capture.
  hipFuncSetAttribute((const void*)TensorODE_rk45_kernel,
                      hipFuncAttributeMaxDynamicSharedMemorySize, SMEM_BYTES);

  dim3 grid(BATCH / ROWS);  // 32 blocks, one 16-row ODE tile each
  dim3 block(512);          // 16 waves (wave32)
  TensorODE_rk45_kernel<<<grid, block, SMEM_BYTES, stream>>>(x, P, C, F, out);
}